// GCN_68719477509
// MI455X (gfx1250) — compile-verified
//
#include <hip/hip_runtime.h>

// GCN (2-layer, GCNConv with self-loops + symmetric norm) for MI455X / gfx1250.
//
//  - Dense X@W (100000x64 @ 64x64) via V_WMMA_F32_16X16X4_F32 (full f32
//    precision). W staged in LDS in *fragment-swizzled* order so each lane's
//    B fragment is one contiguous ds_load_b64 (no register shuffling).
//  - Layer-1 ReLU folded into layer-2 GEMM's A-load (template flag).
//  - Aggregation: out[dst] += dinv[src]*dinv[dst] * H[src] with
//    GLOBAL_ATOMIC_ADD_F32. H (25.6 MB) is resident in the 192 MB L2, so the
//    random gather/scatter resolves in L2; HBM only streams the edge list.

#define GCN_NODES  100000
#define GCN_DIM    64
#define GCN_EDGES  1600000

typedef __attribute__((ext_vector_type(2))) float v2f;
typedef __attribute__((ext_vector_type(8))) float v8f;

// ---------------- degree / normalization ----------------

__global__ void k_deg_init(float* __restrict__ deg) {
    int i = blockIdx.x * blockDim.x + threadIdx.x;
    if (i < GCN_NODES) deg[i] = 1.0f;   // self-loop contributes 1
}

__global__ void k_deg_scatter(const int* __restrict__ dst, float* __restrict__ deg) {
    int e = blockIdx.x * blockDim.x + threadIdx.x;
    if (e < GCN_EDGES) atomicAdd(&deg[dst[e]], 1.0f);
}

__global__ void k_dinv(float* __restrict__ deg) {
    int i = blockIdx.x * blockDim.x + threadIdx.x;
    if (i < GCN_NODES) {
        float d = deg[i];
        deg[i] = (d > 0.0f) ? rsqrtf(d) : 0.0f;
    }
}

// ---------------- dense GEMM: H = X @ W  (K = N = 64) ----------------
// One wave computes a 16-row x 64-col output slab with 4 f32 accumulators,
// V_WMMA_F32_16X16X4_F32 over 16 K-steps. RELU_IN applies relu to A on load.

template <bool RELU_IN>
__global__ void __launch_bounds__(256)
k_gemm64_wmma(const float* __restrict__ X, const float* __restrict__ W,
              float* __restrict__ H, int nrows) {
    // W pre-swizzled into B-fragment order:
    //   sWf[kk*128 + t*32 + lane] = { W[ka][t*16+m], W[ka+1][t*16+m] }
    //   with ka = kk*4 + 2*(lane>>4), m = lane&15.
    // Each fragment load is then a single contiguous ds_load_b64.
    __shared__ float2 sWf[16 * 4 * 32];              // 16 KB of 320 KB/WGP
    for (int f = threadIdx.x; f < 16 * 4 * 32; f += blockDim.x) {
        int kk = f >> 7;                 // k-step
        int t  = (f >> 5) & 3;           // column tile
        int l  = f & 31;                 // lane
        int ka = kk * 4 + 2 * (l >> 4);
        int n  = t * 16 + (l & 15);
        sWf[f] = make_float2(W[ka * GCN_DIM + n], W[(ka + 1) * GCN_DIM + n]);
    }
    __syncthreads();

    const int wave    = threadIdx.x >> 5;
    const int lane    = threadIdx.x & 31;
    const int rowTile = blockIdx.x * (blockDim.x >> 5) + wave;
    const int rowBase = rowTile * 16;
    if (rowBase >= nrows) return;                    // uniform per wave: EXEC all-1s

    const int half = lane >> 4;                      // 0: K even pair, 1: K+2 pair
    const int m    = lane & 15;                      // A row / B column within tile

    v8f acc0 = {}, acc1 = {}, acc2 = {}, acc3 = {};
    const float* xrow = X + (size_t)(rowBase + m) * GCN_DIM;

#pragma unroll
    for (int kk = 0; kk < 16; ++kk) {
        const int ka = kk * 4 + 2 * half;
        // A 16x4 f32 fragment: lanes 0-15 K={k0,k0+1}, lanes 16-31 K={k0+2,k0+3}
        v2f a;
        if (RELU_IN) {
            a.x = fmaxf(xrow[ka], 0.0f);
            a.y = fmaxf(xrow[ka + 1], 0.0f);
        } else {
            a.x = xrow[ka];
            a.y = xrow[ka + 1];
        }
        const float2* bf = &sWf[kk * 128 + lane];
        float2 f0 = bf[0], f1 = bf[32], f2 = bf[64], f3 = bf[96];
        v2f b0, b1, b2, b3;
        b0.x = f0.x; b0.y = f0.y;
        b1.x = f1.x; b1.y = f1.y;
        b2.x = f2.x; b2.y = f2.y;
        b3.x = f3.x; b3.y = f3.y;
        acc0 = __builtin_amdgcn_wmma_f32_16x16x4_f32(false, a, false, b0, (short)0, acc0, false, false);
        acc1 = __builtin_amdgcn_wmma_f32_16x16x4_f32(false, a, false, b1, (short)0, acc1, false, false);
        acc2 = __builtin_amdgcn_wmma_f32_16x16x4_f32(false, a, false, b2, (short)0, acc2, false, false);
        acc3 = __builtin_amdgcn_wmma_f32_16x16x4_f32(false, a, false, b3, (short)0, acc3, false, false);
    }

    // C/D layout: VGPR v -> row (v + 8*half), col m (per 16-col tile)
    float* hbase = H + (size_t)rowBase * GCN_DIM;
#pragma unroll
    for (int v = 0; v < 8; ++v) {
        float* o = hbase + (size_t)(v + 8 * half) * GCN_DIM + m;
        o[0]  = acc0[v];
        o[16] = acc1[v];
        o[32] = acc2[v];
        o[48] = acc3[v];
    }
}

// ---------------- aggregation ----------------

// out[i][d] = bias[d] + dinv[i]^2 * H[i][d]   (self-loop + bias, full init)
__global__ void k_agg_init(const float* __restrict__ H, const float* __restrict__ bias,
                           const float* __restrict__ dinv, float* __restrict__ out) {
    int t = blockIdx.x * blockDim.x + threadIdx.x;
    if (t < GCN_NODES * GCN_DIM) {
        int i = t >> 6;
        int d = t & 63;
        float di = dinv[i];
        out[t] = bias[d] + di * di * H[t];
    }
}

// One wave per edge; each lane handles a float2 of the 64-dim feature row.
// Gather H[src] (L2-resident), scale, atomic scatter into out[dst].
__global__ void k_agg_edges(const float* __restrict__ H,
                            const int* __restrict__ src, const int* __restrict__ dst,
                            const float* __restrict__ dinv, float* __restrict__ out) {
    int gid = blockIdx.x * blockDim.x + threadIdx.x;   // < E*32 = 51.2M
    if (gid >= GCN_EDGES * 32) return;
    int e = gid >> 5;
    int l = gid & 31;
    int s = src[e];
    int d = dst[e];
    float nr = dinv[s] * dinv[d];
    const float2* hp = (const float2*)(H + (size_t)s * GCN_DIM);
    float2 h = hp[l];
    float* op = out + (size_t)d * GCN_DIM + 2 * l;
    atomicAdd(op,     nr * h.x);
    atomicAdd(op + 1, nr * h.y);
}

__global__ void k_relu(float* __restrict__ x) {
    int t = blockIdx.x * blockDim.x + threadIdx.x;
    if (t < GCN_NODES * GCN_DIM) x[t] = fmaxf(x[t], 0.0f);
}

// ---------------- launch ----------------

extern "C" void kernel_launch(void* const* d_in, const int* in_sizes, int n_in,
                              void* d_out, int out_size, void* d_ws, size_t ws_size,
                              hipStream_t stream) {
    const float* node_emb = (const float*)d_in[0];   // [100000, 64]
    const float* W1       = (const float*)d_in[1];   // [64, 64]
    const float* b1       = (const float*)d_in[2];   // [64]
    const float* W2       = (const float*)d_in[3];   // [64, 64]
    const float* b2       = (const float*)d_in[4];   // [64]
    const int*   eidx     = (const int*)d_in[5];     // [2, 1600000]
    const int* src = eidx;
    const int* dst = eidx + GCN_EDGES;
    float* out = (float*)d_out;                      // [100000, 64] (split concat == full X)

    // workspace layout
    float* dinv = (float*)d_ws;                      // 100000 floats (padded region)
    float* H    = dinv + 102400;                     // 6.4M floats (25.6 MB)
    float* X1   = H + (size_t)GCN_NODES * GCN_DIM;   // 6.4M floats (25.6 MB)

    const int T = 256;
    const int gNode = (GCN_NODES + T - 1) / T;
    const int gEdge = (GCN_EDGES + T - 1) / T;
    const int gFeat = (GCN_NODES * GCN_DIM + T - 1) / T;
    const int gScat = (GCN_EDGES * 32 + T - 1) / T;
    const int tiles = GCN_NODES / 16;                // 6250
    const int gGemm = (tiles + 7) / 8;               // 8 waves per block

    // normalization: deg -> dinv (in place)
    k_deg_init   <<<gNode, T, 0, stream>>>(dinv);
    k_deg_scatter<<<gEdge, T, 0, stream>>>(dst, dinv);
    k_dinv       <<<gNode, T, 0, stream>>>(dinv);

    // layer 1: X1 = Agg(node_emb @ W1) + b1   (ReLU deferred into layer-2 A-load)
    k_gemm64_wmma<false><<<gGemm, T, 0, stream>>>(node_emb, W1, H, GCN_NODES);
    k_agg_init   <<<gFeat, T, 0, stream>>>(H, b1, dinv, X1);
    k_agg_edges  <<<gScat, T, 0, stream>>>(H, src, dst, dinv, X1);

    // layer 2: out = relu(Agg(relu(X1) @ W2) + b2)
    k_gemm64_wmma<true><<<gGemm, T, 0, stream>>>(X1, W2, H, GCN_NODES);
    k_agg_init   <<<gFeat, T, 0, stream>>>(H, b2, dinv, out);
    k_agg_edges  <<<gScat, T, 0, stream>>>(H, src, dst, dinv, out);
    k_relu       <<<gFeat, T, 0, stream>>>(out);
}